// FM_12610023981504
// MI455X (gfx1250) — compile-verified
//
#include <hip/hip_runtime.h>
#include <stdint.h>

// ---------------------------------------------------------------------------
// FM (first-order + second-order) gather kernel for MI455X (gfx1250).
// Memory-bound embedding gather -> CDNA5 Tensor Data Mover in GATHER mode
// (32-bit row indices, 8 rows / descriptor) DMAs embedding rows into LDS.
// One wave32 per batch row; the LDS reduction is software-pipelined against
// the in-flight TDM gathers via ordered s_wait_tensorcnt N (N = 4..0).
// ---------------------------------------------------------------------------

#define F_FIELDS 39
#define VOCAB    100000
#define EMB_D    64
#define ROW_BYTES (EMB_D * 4)              // 256 B per embedding row
#define ROWS_PAD  40                        // 39 rows, chunked by 8
#define WAVE_BUF  (ROWS_PAD * ROW_BYTES)    // 10240 B LDS per wave
#define WAVES_PER_BLOCK 4
#define BLOCK_THREADS   (WAVES_PER_BLOCK * 32)

typedef __attribute__((ext_vector_type(4))) unsigned int v4u;
typedef __attribute__((ext_vector_type(4))) int          v4i;
typedef __attribute__((ext_vector_type(8))) int          v8i;

#if defined(__HIP_DEVICE_COMPILE__) && __has_builtin(__builtin_amdgcn_tensor_load_to_lds)
#define FM_HAS_TDM 1
#else
#define FM_HAS_TDM 0
#endif

#if FM_HAS_TDM
// Wait count must be an integral constant expression at Sema time -> template.
template <int C, int NROWS>
__device__ __forceinline__ void fm_chunk(const float2* __restrict__ buf, int lane,
                                         float& s0, float& s1, float& sq)
{
    __builtin_amdgcn_s_wait_tensorcnt(4 - C);   // chunk C landed (in-order TDM)
#pragma unroll
    for (int j = 0; j < NROWS; ++j) {
        const int f = C * 8 + j;
        float2 v = buf[f * 32 + lane];
        s0 += v.x;
        s1 += v.y;
        sq += v.x * v.x + v.y * v.y;
    }
}
#endif

extern "C" __global__ __launch_bounds__(BLOCK_THREADS)
void fm_gather_kernel(const int* __restrict__ Xi,
                      const float* __restrict__ W1,
                      const float* __restrict__ W2,
                      const float* __restrict__ bias,
                      float* __restrict__ out,
                      int B)
{
    extern __shared__ char smem[];

    const int lane = (int)(threadIdx.x & 31u);
    const int wave = (int)(threadIdx.x >> 5u);
    const int b    = (int)blockIdx.x * WAVES_PER_BLOCK + wave;
    if (b >= B) return;

    // ---- load indices: f = lane (0..31) and f = 32+lane (lanes 0..6) ----
    const int base = b * F_FIELDS;
    int rA = Xi[base + lane] + lane * VOCAB;                 // rows f = 0..31
    int rB = 0;
    const bool hasB = (lane < F_FIELDS - 32);                // lanes 0..6
    if (hasB) rB = Xi[base + 32 + lane] + (32 + lane) * VOCAB;

    // ---- first order: scalar gathers from W1 ----
    float linp = W1[rA] + (hasB ? W1[rB] : 0.0f);

    const int idx38 = __shfl(rB, F_FIELDS - 32 - 1);         // last valid row

    float s0 = 0.0f, s1 = 0.0f, sq = 0.0f;

#if FM_HAS_TDM
    // ---- TDM gather descriptors: 5 ops x 8 rows x 256 B into LDS ----
    const unsigned lds_base = (unsigned)(wave * WAVE_BUF);
    const uint64_t gaddr = (uint64_t)(uintptr_t)W2;

    v4u g0c;
    g0c[0] = 0xC0000001u;                      // count=1 | gather_mode | 32-bit idx
    g0c[1] = 0u;                               // lds_addr (per chunk)
    g0c[2] = (unsigned)(gaddr & 0xFFFFFFFFu);  // global_addr[31:0]
    g0c[3] = (unsigned)((gaddr >> 32) & 0x01FFFFFFu) | 0x80000000u; // addr[56:32] | type=2

    const unsigned TD1 = (unsigned)(F_FIELDS * VOCAB);       // 3,900,000 rows
    v8i g1c;
    g1c[0] = (int)(2u << 16);                                // data_size = 4 B
    g1c[1] = (int)(((unsigned)EMB_D & 0xFFFFu) << 16);       // tensor_dim0 lo16
    g1c[2] = (int)(((TD1 & 0xFFFFu) << 16) |
                   (((unsigned)EMB_D >> 16) & 0xFFFFu));     // dim1 lo16 | dim0 hi16
    g1c[3] = (int)(((unsigned)EMB_D << 16) |
                   ((TD1 >> 16) & 0xFFFFu));                 // tile_dim0 | dim1 hi16
    g1c[4] = 8;                                              // tile_dim1 = #indices
    g1c[5] = EMB_D;                                          // dim0_stride lo32 = 64
    g1c[6] = 0;
    g1c[7] = 0;

#pragma unroll
    for (int c = 0; c < 5; ++c) {
        int ri[8];
#pragma unroll
        for (int j = 0; j < 8; ++j) {
            const int f = c * 8 + j;
            int r;
            if (f < 32)              r = __shfl(rA, f);
            else if (f < F_FIELDS)   r = __shfl(rB, f - 32);
            else                     r = idx38;              // padding (repeat ok)
            ri[j] = r;
        }
        v4u G0 = g0c;
        G0[1] = lds_base + (unsigned)(c * 8 * ROW_BYTES);
        v8i G1 = g1c;
        G1[4] = (c == 4) ? (F_FIELDS - 32) : 8;              // 7 valid on last chunk
        v4i G2; G2[0] = ri[0]; G2[1] = ri[1]; G2[2] = ri[2]; G2[3] = ri[3];
        v4i G3; G3[0] = ri[4]; G3[1] = ri[5]; G3[2] = ri[6]; G3[3] = ri[7];
#if __clang_major__ >= 23
        v8i GZ; GZ[0]=0; GZ[1]=0; GZ[2]=0; GZ[3]=0; GZ[4]=0; GZ[5]=0; GZ[6]=0; GZ[7]=0;
        __builtin_amdgcn_tensor_load_to_lds(G0, G1, G2, G3, GZ, 0);
#else
        __builtin_amdgcn_tensor_load_to_lds(G0, G1, G2, G3, 0);
#endif
    }

    // ---- pipelined reduction: TDM ops complete in order per wave, so after
    // issuing 5, tensorcnt<=4-c guarantees chunk c has landed in LDS. ----
    const float2* buf = (const float2*)(smem + wave * WAVE_BUF);
    fm_chunk<0, 8>(buf, lane, s0, s1, sq);
    fm_chunk<1, 8>(buf, lane, s0, s1, sq);
    fm_chunk<2, 8>(buf, lane, s0, s1, sq);
    fm_chunk<3, 8>(buf, lane, s0, s1, sq);
    fm_chunk<4, F_FIELDS - 32>(buf, lane, s0, s1, sq);       // 7 rows
#else
    // ---- fallback: direct per-lane float2 gathers from global ----
#pragma unroll 13
    for (int f = 0; f < F_FIELDS; ++f) {
        int r = (f < 32) ? __shfl(rA, f) : __shfl(rB, f - 32);
        const float2* row = (const float2*)(W2 + (size_t)r * EMB_D);
        float2 v = row[lane];
        s0 += v.x;
        s1 += v.y;
        sq += v.x * v.x + v.y * v.y;
    }
#endif

    // second order contribution for this lane's two dims
    float total = 0.5f * (s0 * s0 + s1 * s1 - sq) + linp;

    // wave32 butterfly reduction
#pragma unroll
    for (int off = 16; off > 0; off >>= 1)
        total += __shfl_xor(total, off);

    if (lane == 0)
        out[b] = total + bias[0];
}

extern "C" void kernel_launch(void* const* d_in, const int* in_sizes, int n_in,
                              void* d_out, int out_size, void* d_ws, size_t ws_size,
                              hipStream_t stream) {
    const int*   Xi   = (const int*)d_in[0];     // [B, F, 1] int32
    const float* W1   = (const float*)d_in[1];   // [F*VOCAB, 1]
    const float* W2   = (const float*)d_in[2];   // [F*VOCAB, D]
    const float* bias = (const float*)d_in[3];   // [1]
    float*       out  = (float*)d_out;           // [B]

    const int B = in_sizes[0] / F_FIELDS;
    const int blocks = (B + WAVES_PER_BLOCK - 1) / WAVES_PER_BLOCK;
    const size_t shmem = (size_t)WAVES_PER_BLOCK * WAVE_BUF;  // 40 KB

    fm_gather_kernel<<<blocks, BLOCK_THREADS, shmem, stream>>>(Xi, W1, W2, bias, out, B);
}